// SSKernelDiag_60215441490627
// MI455X (gfx1250) — compile-verified
//
#include <hip/hip_runtime.h>

typedef __attribute__((ext_vector_type(2))) float v2f;
typedef __attribute__((ext_vector_type(8))) float v8f;

namespace {
constexpr int kH = 1024;   // heads
constexpr int kN = 64;     // state size (contraction dim)
constexpr int kL = 2048;   // sequence length = 128 * 16

struct cpx { float r, i; };
__device__ __forceinline__ cpx cmul(cpx a, cpx b) {
  return { fmaf(a.r, b.r, -a.i * b.i), fmaf(a.r, b.i, a.i * b.r) };
}
} // namespace

// One workgroup per h. K[h, 16*i + j] = 2*Re( sum_n Ct[n] * dA16[n]^i * dA[n]^j )
//                                     = 2*( Wr^T Qr  -  Wi^T Qi )   (two real f32 GEMMs)
__global__ __launch_bounds__(256)
void ssd_vandermonde_wmma(const float* __restrict__ C_ri,
                          const float* __restrict__ log_dt,
                          const float* __restrict__ B_ri,
                          const float* __restrict__ inv_A_real,
                          const float* __restrict__ A_imag,
                          float* __restrict__ K_out) {
  __shared__ float sWr[kN][65];   // Re(Ct * dA16^i), i-half of 64, padded
  __shared__ float sWi[kN][65];   // -Im(Ct * dA16^i)  (negated: f32 WMMA has no A/B neg)
  __shared__ float sQr[kN][17];   // Re(dA^j), j<16, padded
  __shared__ float sQi[kN][17];   // Im(dA^j)
  __shared__ float sCtr[kN], sCti[kN];
  __shared__ float s16r[kN], s16i[kN];     // dA^16
  __shared__ float s256r[kN], s256i[kN];   // dA^256

  const int h   = blockIdx.x;
  const int tid = threadIdx.x;

  // ---- Stage 1: per-n discretization + small power tables (64 lanes) ----
  if (tid < kN) {
    const int n   = tid;
    const int idx = h * kN + n;                 // rep == 1: A/B indexed by h directly
    const float dt = expf(log_dt[h]);
    const cpx  A   = { -expf(inv_A_real[idx]), A_imag[idx] };
    const cpx  dtA = { A.r * dt, A.i * dt };
    const cpx  den = { 1.0f - 0.5f * dtA.r, -0.5f * dtA.i };   // 1 - dtA/2
    const cpx  num = { 1.0f + 0.5f * dtA.r,  0.5f * dtA.i };   // 1 + dtA/2
    const float inv = 1.0f / fmaf(den.r, den.r, den.i * den.i);
    const cpx  dA  = { (num.r * den.r + num.i * den.i) * inv,
                       (num.i * den.r - num.r * den.i) * inv };
    const cpx  C   = { C_ri[2 * idx], C_ri[2 * idx + 1] };     // CH == 1
    const cpx  B   = { B_ri[2 * idx], B_ri[2 * idx + 1] };
    const cpx  BC  = cmul(B, C);
    const cpx  Ct  = { (BC.r * den.r + BC.i * den.i) * inv * dt,
                       (BC.i * den.r - BC.r * den.i) * inv * dt };

    // Q[j] = dA^j for j = 0..15 (geometric recurrence); loop exits with q = dA^16
    cpx q = { 1.0f, 0.0f };
    #pragma unroll
    for (int j = 0; j < 16; ++j) { sQr[n][j] = q.r; sQi[n][j] = q.i; q = cmul(q, dA); }
    const cpx dA16 = q;
    cpx t = dA16;                                   // dA16^16 = dA^256 (4 squarings)
    t = cmul(t, t); t = cmul(t, t); t = cmul(t, t); t = cmul(t, t);

    sCtr[n] = Ct.r;  sCti[n] = Ct.i;
    s16r[n] = dA16.r; s16i[n] = dA16.i;
    s256r[n] = t.r;   s256i[n] = t.i;
  }

  const int wave = tid >> 5;       // wave32
  const int lane = tid & 31;
  const int m    = lane & 15;      // N-index / row-in-half
  const int hi   = lane >> 4;      // lane half
  float* outh = K_out + (size_t)h * kL;

  // ---- Two passes over i-halves: i_global = 64*pass + i_local, i_local in [0,64) ----
  for (int pass = 0; pass < 2; ++pass) {
    __syncthreads();   // stage-1 done / previous pass's WMMA reads done

    // Fill W half: 256 threads, thread -> (n = tid&63, group g = tid>>6 of 16 i-values)
    {
      const int n = tid & 63;
      const int g = tid >> 6;             // 0..3
      const int s = 4 * pass + g;         // start exponent in dA256 units (0..7)
      const cpx b256 = { s256r[n], s256i[n] };
      cpx e = { 1.0f, 0.0f };
      for (int k = 0; k < s; ++k) e = cmul(e, b256);       // wave-uniform trip count
      cpx P = cmul(cpx{ sCtr[n], sCti[n] }, e);            // Ct * dA^(256*s)
      const cpx d16 = { s16r[n], s16i[n] };
      #pragma unroll
      for (int k = 0; k < 16; ++k) {
        const int il = 16 * g + k;
        sWr[n][il] = P.r;
        sWi[n][il] = -P.i;
        P = cmul(P, d16);
      }
    }
    __syncthreads();

    // WMMA: waves 0..3 each own one 16x16 tile of this i-half (wave-uniform branch)
    if (wave < 4) {
      const int i0 = 16 * wave;          // local row base within the half
      v8f acc = {};
      // Real part:  acc += Wr_tile^T (16x4 chunks) * Qr (4x16)
      #pragma unroll
      for (int kc = 0; kc < 16; ++kc) {
        const int n0 = 4 * kc + 2 * hi;
        v2f a, b;
        a[0] = sWr[n0][i0 + m];  a[1] = sWr[n0 + 1][i0 + m];   // A: M=m, K=2*hi, 2*hi+1
        b[0] = sQr[n0][m];       b[1] = sQr[n0 + 1][m];        // B: K=2*hi(+1), N=m
        acc = __builtin_amdgcn_wmma_f32_16x16x4_f32(false, a, false, b,
                                                    (short)0, acc, false, false);
      }
      // Imag part: acc += (-Wi_tile)^T * Qi   (negation folded into sWi)
      #pragma unroll
      for (int kc = 0; kc < 16; ++kc) {
        const int n0 = 4 * kc + 2 * hi;
        v2f a, b;
        a[0] = sWi[n0][i0 + m];  a[1] = sWi[n0 + 1][i0 + m];
        b[0] = sQi[n0][m];       b[1] = sQi[n0 + 1][m];
        acc = __builtin_amdgcn_wmma_f32_16x16x4_f32(false, a, false, b,
                                                    (short)0, acc, false, false);
      }
      // Store: D layout — VGPR r: lanes 0-15 M=r, lanes 16-31 M=r+8; N = lane%16
      const int i_base = 64 * pass + i0;
      #pragma unroll
      for (int r = 0; r < 8; ++r) {
        const int M = r + 8 * hi;
        outh[(i_base + M) * 16 + m] = 2.0f * acc[r];
      }
    }
  }
}

extern "C" void kernel_launch(void* const* d_in, const int* in_sizes, int n_in,
                              void* d_out, int out_size, void* d_ws, size_t ws_size,
                              hipStream_t stream) {
  (void)in_sizes; (void)n_in; (void)out_size; (void)d_ws; (void)ws_size;
  const float* C_ri       = (const float*)d_in[0];
  const float* log_dt     = (const float*)d_in[1];
  const float* B_ri       = (const float*)d_in[2];
  const float* inv_A_real = (const float*)d_in[3];
  const float* A_imag     = (const float*)d_in[4];
  // d_in[5] is L (python int scalar) — compile-time constant kL = 2048 here.
  float* out = (float*)d_out;

  ssd_vandermonde_wmma<<<dim3(kH), dim3(256), 0, stream>>>(
      C_ri, log_dt, B_ri, inv_A_real, A_imag, out);
}